// _x_to_y_74071005986822
// MI455X (gfx1250) — compile-verified
//
#include <hip/hip_runtime.h>

typedef __attribute__((ext_vector_type(2))) float v2f;
typedef __attribute__((ext_vector_type(8))) float v8f;

#define WAVES_PER_BLOCK 8
#define BLOCK_THREADS   (WAVES_PER_BLOCK * 32)
#define LDS_STRIDE      20   // even -> 8B-aligned b64 reads; padded vs 16 to dodge bank conflicts

#define LOG2E 1.44269504088896340736f
#define LN2   0.69314718055994530942f

__device__ __forceinline__ float softplus_f(float x) {
    // softplus(x) = max(x,0) + log1p(exp(-|x|))
    //             = max(x,0) + ln2 * log2(1 + 2^(-|x|*log2e))
    // Argument of log2 is in (1,2] -> never denormal, so raw v_exp_f32/v_log_f32
    // builtins are exact enough and skip libm's denormal-rescue fixups.
    float t = __builtin_amdgcn_exp2f(-fabsf(x) * LOG2E);   // v_exp_f32 (abs+mul folded)
    float l = __builtin_amdgcn_logf(1.0f + t);             // v_log_f32 (base-2)
    return fmaf(LN2, l, fmaxf(x, 0.0f));
}

__global__ __launch_bounds__(BLOCK_THREADS) void hyperelastic_mlp_kernel(
    const float* __restrict__ Fs,
    const float* __restrict__ W1, const float* __restrict__ b1,
    const float* __restrict__ W2, const float* __restrict__ b2,
    const float* __restrict__ W3, const float* __restrict__ b3,
    const float* __restrict__ W4, const float* __restrict__ b4,
    float* __restrict__ out, int n_samples)
{
    __shared__ float lds[WAVES_PER_BLOCK][16 * LDS_STRIDE];

    const int lane = threadIdx.x & 31;
    const int wid  = threadIdx.x >> 5;
    const int m    = lane & 15;  // A-row (sample in tile) == B/D column index
    const int half = lane >> 4;  // selects K pair {2*half, 2*half+1} within each K=4 step
    float* my_lds = lds[wid];

    // ---- per-wave weight registers in WMMA B layout (4x16 f32: V0=row 2h, V1=row 2h+1) ----
    v2f w1r[5], w2r[4], w3r[4];
    #pragma unroll
    for (int s = 0; s < 5; ++s) {
        int r0 = 4 * s + 2 * half;
        w1r[s].x = (r0     < 19) ? W1[r0 * 16 + m]       : 0.0f;
        w1r[s].y = (r0 + 1 < 19) ? W1[(r0 + 1) * 16 + m] : 0.0f;
    }
    #pragma unroll
    for (int s = 0; s < 4; ++s) {
        int r0 = 4 * s + 2 * half;
        w2r[s].x = W2[r0 * 16 + m];  w2r[s].y = W2[(r0 + 1) * 16 + m];
        w3r[s].x = W3[r0 * 16 + m];  w3r[s].y = W3[(r0 + 1) * 16 + m];
    }
    const float b1v = b1[m], b2v = b2[m], b3v = b3[m];
    const float w4v = W4[m];
    const float b4v = b4[0];
    // bias as WMMA C initializer (element (M,N) needs bias[N], uniform per lane)
    v8f cb1, cb2, cb3;
    #pragma unroll
    for (int v = 0; v < 8; ++v) { cb1[v] = b1v; cb2[v] = b2v; cb3[v] = b3v; }

    const int tiles      = (n_samples + 15) >> 4;
    const int waveGlobal = blockIdx.x * WAVES_PER_BLOCK + wid;
    const int waveStride = gridDim.x * WAVES_PER_BLOCK;

    for (int tile = waveGlobal; tile < tiles; tile += waveStride) {
        int sample = tile * 16 + m;
        if (sample >= n_samples) sample = n_samples - 1;  // clamp (tail tile only)
        const float* Fp = Fs + (size_t)sample * 9;

        // ---- features: F(9), cofactor(9), det, pad ----
        float a0 = Fp[0], a1 = Fp[1], a2 = Fp[2];
        float a3 = Fp[3], a4 = Fp[4], a5 = Fp[5];
        float a6 = Fp[6], a7 = Fp[7], a8 = Fp[8];
        float c00 = a4 * a8 - a5 * a7;
        float c01 = a5 * a6 - a3 * a8;
        float c02 = a3 * a7 - a4 * a6;
        float c10 = a2 * a7 - a1 * a8;
        float c11 = a0 * a8 - a2 * a6;
        float c12 = a1 * a6 - a0 * a7;
        float c20 = a1 * a5 - a2 * a4;
        float c21 = a2 * a3 - a0 * a5;
        float c22 = a0 * a4 - a1 * a3;
        float det = a0 * c00 + a1 * c01 + a2 * c02;
        float f[20] = {a0, a1, a2, a3, a4, a5, a6, a7, a8,
                       c00, c01, c02, c10, c11, c12, c20, c21, c22, det, 0.0f};

        // ---- layer 1: x(16x20) @ W1(20x16) via 5x V_WMMA_F32_16X16X4_F32, C0 = bias ----
        v8f c = cb1;
        #pragma unroll
        for (int s = 0; s < 5; ++s) {
            const int ks = 4 * s;
            v2f a;
            a.x = half ? f[ks + 2] : f[ks + 0];   // constant indices -> stays in VGPRs
            a.y = half ? f[ks + 3] : f[ks + 1];
            c = __builtin_amdgcn_wmma_f32_16x16x4_f32(false, a, false, w1r[s],
                                                      (short)0, c, false, false);
        }
        #pragma unroll
        for (int v = 0; v < 8; ++v) c[v] = softplus_f(c[v]);

        // ---- layer 2: D-layout -> LDS (row-major 16x16) -> A-layout ----
        #pragma unroll
        for (int v = 0; v < 8; ++v)
            my_lds[(v + 8 * half) * LDS_STRIDE + m] = c[v];
        v8f c2 = cb2;
        #pragma unroll
        for (int s = 0; s < 4; ++s) {
            const int k0 = 4 * s + 2 * half;
            v2f a;
            a.x = my_lds[m * LDS_STRIDE + k0];
            a.y = my_lds[m * LDS_STRIDE + k0 + 1];
            c2 = __builtin_amdgcn_wmma_f32_16x16x4_f32(false, a, false, w2r[s],
                                                       (short)0, c2, false, false);
        }
        #pragma unroll
        for (int v = 0; v < 8; ++v) c2[v] = softplus_f(c2[v]);

        // ---- layer 3 ----
        #pragma unroll
        for (int v = 0; v < 8; ++v)
            my_lds[(v + 8 * half) * LDS_STRIDE + m] = c2[v];
        v8f c3 = cb3;
        #pragma unroll
        for (int s = 0; s < 4; ++s) {
            const int k0 = 4 * s + 2 * half;
            v2f a;
            a.x = my_lds[m * LDS_STRIDE + k0];
            a.y = my_lds[m * LDS_STRIDE + k0 + 1];
            c3 = __builtin_amdgcn_wmma_f32_16x16x4_f32(false, a, false, w3r[s],
                                                       (short)0, c3, false, false);
        }
        #pragma unroll
        for (int v = 0; v < 8; ++v) c3[v] = softplus_f(c3[v]);

        // ---- layer 4: out[M] = sum_n h3[M][n] * W4[n] + b4 via xor-reduction over 16 lanes ----
        float r[8];
        #pragma unroll
        for (int v = 0; v < 8; ++v) r[v] = c3[v] * w4v;
        #pragma unroll
        for (int mask = 1; mask <= 8; mask <<= 1) {
            #pragma unroll
            for (int v = 0; v < 8; ++v) r[v] += __shfl_xor(r[v], mask, 32);
        }
        // lanes 0-15: r[v] = out[v]; lanes 16-31: r[v] = out[v+8]

        float* op = out + (size_t)tile * 16;
        if (tile * 16 + 16 <= n_samples) {
            if (lane == 0) {
                *reinterpret_cast<float4*>(op + 0) =
                    make_float4(r[0] + b4v, r[1] + b4v, r[2] + b4v, r[3] + b4v);
                *reinterpret_cast<float4*>(op + 4) =
                    make_float4(r[4] + b4v, r[5] + b4v, r[6] + b4v, r[7] + b4v);
            } else if (lane == 16) {
                *reinterpret_cast<float4*>(op + 8) =
                    make_float4(r[0] + b4v, r[1] + b4v, r[2] + b4v, r[3] + b4v);
                *reinterpret_cast<float4*>(op + 12) =
                    make_float4(r[4] + b4v, r[5] + b4v, r[6] + b4v, r[7] + b4v);
            }
        } else {  // tail tile: guarded scalar stores
            if (lane == 0) {
                #pragma unroll
                for (int v = 0; v < 8; ++v)
                    if (tile * 16 + v < n_samples) op[v] = r[v] + b4v;
            } else if (lane == 16) {
                #pragma unroll
                for (int v = 0; v < 8; ++v)
                    if (tile * 16 + 8 + v < n_samples) op[8 + v] = r[v] + b4v;
            }
        }
    }
}

extern "C" void kernel_launch(void* const* d_in, const int* in_sizes, int n_in,
                              void* d_out, int out_size, void* d_ws, size_t ws_size,
                              hipStream_t stream) {
    const float* Fs = (const float*)d_in[0];
    const float* W1 = (const float*)d_in[1];
    const float* b1 = (const float*)d_in[2];
    const float* W2 = (const float*)d_in[3];
    const float* b2 = (const float*)d_in[4];
    const float* W3 = (const float*)d_in[5];
    const float* b3 = (const float*)d_in[6];
    const float* W4 = (const float*)d_in[7];
    const float* b4 = (const float*)d_in[8];
    float* out = (float*)d_out;

    const int n = in_sizes[0] / 9;          // N samples (Fs is N*3*3)
    const int tiles = (n + 15) >> 4;
    int blocks = 2048;                      // persistent grid-stride; amortizes weight loads
    const int maxBlocks = (tiles + WAVES_PER_BLOCK - 1) / WAVES_PER_BLOCK;
    if (blocks > maxBlocks) blocks = maxBlocks;
    if (blocks < 1) blocks = 1;

    hyperelastic_mlp_kernel<<<blocks, BLOCK_THREADS, 0, stream>>>(
        Fs, W1, b1, W2, b2, W3, b3, W4, b4, out, n);
}